// MambaLayer_48473000903653
// MI455X (gfx1250) — compile-verified
//
#include <hip/hip_runtime.h>
#include <math.h>

// ---- Problem constants (match the reference) ----
#define DIM      768
#define D_INNER  1536
#define D_STATE  16
#define D_CONV   4
#define DT_RANK  48
#define BATCH    2
#define SEQ      1024
#define NTOK     (BATCH * SEQ)            // 2048 tokens
#define DBC_W    (DT_RANK + 2 * D_STATE)  // 80

typedef float v2f __attribute__((ext_vector_type(2)));
typedef float v8f __attribute__((ext_vector_type(8)));

// ---------------------------------------------------------------------------
// CDNA5 async global->LDS helpers (gfx1250). LDS byte address = low 32 bits of
// the generic pointer. Tracked by
// ASYNCcnt; drained with s_wait_asynccnt.
// ---------------------------------------------------------------------------
__device__ __forceinline__ unsigned lds_off(const void* p) {
    return (unsigned)(uintptr_t)p;
}
__device__ __forceinline__ void async_copy_8B(unsigned lds, const float* g) {
    asm volatile("global_load_async_to_lds_b64 %0, %1, off"
                 :: "v"(lds), "v"(g) : "memory");
}
__device__ __forceinline__ void async_wait0() {
    asm volatile("s_wait_asynccnt 0" ::: "memory");
}

// ---------------------------------------------------------------------------
// Kernel 1: LayerNorm. One wave (32 lanes) per token row; wave32 shuffles.
// ---------------------------------------------------------------------------
__global__ __launch_bounds__(256) void mamba_ln_kernel(
    const float* __restrict__ x, const float* __restrict__ w,
    const float* __restrict__ b, float* __restrict__ xn)
{
    const int lane = threadIdx.x & 31;
    const int row  = blockIdx.x * 8 + (threadIdx.x >> 5);   // 2048 rows
    const float* xr = x + (size_t)row * DIM;

    float s = 0.f;
    for (int j = lane; j < DIM; j += 32) s += xr[j];
#pragma unroll
    for (int o = 16; o > 0; o >>= 1) s += __shfl_xor(s, o, 32);
    const float mu = s * (1.0f / DIM);

    float v = 0.f;
    for (int j = lane; j < DIM; j += 32) { float d = xr[j] - mu; v += d * d; }
#pragma unroll
    for (int o = 16; o > 0; o >>= 1) v += __shfl_xor(v, o, 32);
    const float rstd = rsqrtf(v * (1.0f / DIM) + 1e-5f);

    float* o = xn + (size_t)row * DIM;
    for (int j = lane; j < DIM; j += 32)
        o[j] = (xr[j] - mu) * rstd * w[j] + b[j];
}

// ---------------------------------------------------------------------------
// Generic fp32 WMMA GEMM:  C[M,N] = A[M,K(lda)] * W[N,K]^T  (+ epilogue)
//   Block: 256 threads = 8 wave32s arranged 2(M) x 4(N) -> 32x64 output tile.
//   Double-buffered K stages of 16, staged via global_load_async_to_lds_b64,
//   consumed by 4 x v_wmma_f32_16x16x4_f32 per wave per stage.
//   EPI==1: v = softplus(v + bias[n]).
// Fragment layouts per CDNA5 ISA 7.12.2:
//   A 16x4:  lanes 0-15 -> M=lane, K={0,1} in v[0],v[1]; lanes 16-31 -> K={2,3}
//   B 4x16:  lanes 0-15 -> N=lane, K={0,1};              lanes 16-31 -> K={2,3}
//   C/D:     VGPR i: lanes 0-15 -> M=i, N=lane; lanes 16-31 -> M=i+8, N=lane-16
// LDS layouts: sA[buf][m][k] (stride 20), sB[buf][n][k] (stride 20) so each
// 8-byte async chunk is LDS-contiguous and fragment reads are conflict-free.
// ---------------------------------------------------------------------------
template <int EPI>
__global__ __launch_bounds__(256) void mamba_gemm_wmma_f32(
    const float* __restrict__ A, int lda,
    const float* __restrict__ W,
    const float* __restrict__ bias,
    float* __restrict__ C,
    int M, int N, int K)
{
    __shared__ float sA[2][32][20];   // 2 x 32 rows x 16 K (+4 pad)
    __shared__ float sB[2][64][20];   // 2 x 64 cols x 16 K (+4 pad)

    const int tid  = threadIdx.x;
    const int lane = tid & 31;
    const int wave = tid >> 5;
    const int wm   = wave & 1;        // 0..1  (M tile within block)
    const int wn   = wave >> 1;       // 0..3  (N tile within block)
    const int row0 = blockIdx.x * 32;
    const int col0 = blockIdx.y * 64;

    v8f acc = {0.f, 0.f, 0.f, 0.f, 0.f, 0.f, 0.f, 0.f};

    // fragment read positions in LDS
    const int aM = wm * 16 + (lane & 15);
    const int aK = (lane >> 4) << 1;          // 0 or 2
    const int bN = wn * 16 + (lane & 15);
    const int bK = aK;

    // cooperative staging positions (256 threads x 8 bytes per tile pass)
    const int lr  = tid >> 3;                 // 0..31 (A row / B col base)
    const int lc  = (tid & 7) << 1;           // 0..14 (K pair)

    const int nstages = K >> 4;               // K % 16 == 0 for all uses

    auto stage = [&](int s, int buf) {
        const int k0 = s << 4;
        // A tile: 32 x 16 floats, always in range (M multiple of 32)
        async_copy_8B(lds_off(&sA[buf][lr][lc]),
                      A + (size_t)(row0 + lr) * lda + k0 + lc);
        // B tile: 64 x 16 floats from row-major W[N][K]
#pragma unroll
        for (int j = 0; j < 2; ++j) {
            const int n  = lr + 32 * j;
            const int gn = col0 + n;
            if (gn < N) {
                async_copy_8B(lds_off(&sB[buf][n][lc]),
                              W + (size_t)gn * K + k0 + lc);
            } else {                           // zero-fill OOB columns
                sB[buf][n][lc]     = 0.f;
                sB[buf][n][lc + 1] = 0.f;
            }
        }
    };

    stage(0, 0);
    async_wait0();
    __syncthreads();

    for (int s = 0; s < nstages; ++s) {
        const int cur = s & 1;
        if (s + 1 < nstages) stage(s + 1, cur ^ 1);   // prefetch next stage

#pragma unroll
        for (int kk = 0; kk < 16; kk += 4) {
            v2f a, b;
            a.x = sA[cur][aM][kk + aK];
            a.y = sA[cur][aM][kk + aK + 1];
            b.x = sB[cur][bN][kk + bK];
            b.y = sB[cur][bN][kk + bK + 1];
            acc = __builtin_amdgcn_wmma_f32_16x16x4_f32(
                false, a, false, b, (short)0, acc, false, false);
        }
        async_wait0();        // next-stage deposits complete
        __syncthreads();      // all waves done reading cur before overwrite
    }

    // epilogue + store (strided per C/D layout)
    const int cn    = col0 + wn * 16 + (lane & 15);
    const int rbase = row0 + wm * 16 + ((lane >> 4) << 3);
    if (cn < N) {
        float bv = 0.f;
        if (EPI == 1) bv = bias[cn];
#pragma unroll
        for (int i = 0; i < 8; ++i) {
            float v = acc[i];
            if (EPI == 1) {            // softplus(v + bias)
                v += bv;
                v = (v > 20.f) ? v : log1pf(__expf(v));
            }
            C[(size_t)(rbase + i) * N + cn] = v;
        }
    }
}

// ---------------------------------------------------------------------------
// Kernel 3: causal depthwise conv1d (k=4) + SiLU. One thread per (b,l,d).
// Reads x-branch (first D_INNER cols) of xz.
// ---------------------------------------------------------------------------
__global__ __launch_bounds__(256) void mamba_conv_silu_kernel(
    const float* __restrict__ xz, const float* __restrict__ cw,
    const float* __restrict__ cb, float* __restrict__ u)
{
    const int idx = blockIdx.x * 256 + threadIdx.x;   // < NTOK * D_INNER
    const int d   = idx % D_INNER;
    const int l   = (idx / D_INNER) % SEQ;
    const int b   = idx / (D_INNER * SEQ);

    const float* base = xz + (size_t)b * SEQ * (2 * D_INNER);
    float acc = cb[d];
#pragma unroll
    for (int k = 0; k < D_CONV; ++k) {
        const int ls = l + k - (D_CONV - 1);
        if (ls >= 0)
            acc += base[(size_t)ls * (2 * D_INNER) + d] * cw[d * D_CONV + k];
    }
    u[(size_t)idx] = acc / (1.f + __expf(-acc));      // SiLU
}

// ---------------------------------------------------------------------------
// Kernel 6: selective scan. One thread per (b,d), 16-wide state in registers.
// Per-timestep B/C vectors (shared across d) broadcast via LDS.
// Fuses the D skip-connection and the SiLU(z) gate.
// ---------------------------------------------------------------------------
__global__ __launch_bounds__(256) void mamba_scan_kernel(
    const float* __restrict__ u, const float* __restrict__ delta,
    const float* __restrict__ dbc, const float* __restrict__ xz,
    const float* __restrict__ A_log, const float* __restrict__ Dp,
    float* __restrict__ y)
{
    __shared__ float sBC[32];    // [0..15]=B_t, [16..31]=C_t
    const int tid = threadIdx.x;
    const int b   = blockIdx.x / (D_INNER / 256);
    const int d   = (blockIdx.x % (D_INNER / 256)) * 256 + tid;

    float a[D_STATE], h[D_STATE];
#pragma unroll
    for (int n = 0; n < D_STATE; ++n) {
        a[n] = -__expf(A_log[d * D_STATE + n]);
        h[n] = 0.f;
    }
    const float Dd = Dp[d];

    for (int t = 0; t < SEQ; ++t) {
        const size_t tok = (size_t)b * SEQ + t;
        if (tid < 32)   // B,C are contiguous cols 48..79 of dbc
            sBC[tid] = dbc[tok * DBC_W + DT_RANK + tid];
        __syncthreads();

        const float dlt = delta[tok * D_INNER + d];
        const float uu  = u[tok * D_INNER + d];
        const float du  = dlt * uu;
        float yv = 0.f;
#pragma unroll
        for (int n = 0; n < D_STATE; ++n) {
            h[n] = __expf(dlt * a[n]) * h[n] + du * sBC[n];
            yv  += h[n] * sBC[16 + n];
        }
        yv += uu * Dd;
        const float zv = xz[tok * (2 * D_INNER) + D_INNER + d];
        yv *= zv / (1.f + __expf(-zv));               // * SiLU(z)
        y[tok * D_INNER + d] = yv;
        __syncthreads();
    }
}

// ---------------------------------------------------------------------------
extern "C" void kernel_launch(void* const* d_in, const int* in_sizes, int n_in,
                              void* d_out, int out_size, void* d_ws, size_t ws_size,
                              hipStream_t stream)
{
    const float* x      = (const float*)d_in[0];
    const float* ln_w   = (const float*)d_in[1];
    const float* ln_b   = (const float*)d_in[2];
    const float* in_w   = (const float*)d_in[3];   // (3072, 768)
    const float* conv_w = (const float*)d_in[4];   // (1536, 4)
    const float* conv_b = (const float*)d_in[5];
    const float* xp_w   = (const float*)d_in[6];   // (80, 1536)
    const float* dt_w   = (const float*)d_in[7];   // (1536, 48)
    const float* dt_b   = (const float*)d_in[8];
    const float* A_log  = (const float*)d_in[9];   // (1536, 16)
    const float* Dp     = (const float*)d_in[10];
    const float* out_w  = (const float*)d_in[11];  // (768, 1536)
    float* out = (float*)d_out;

    float* ws    = (float*)d_ws;
    float* xn    = ws;  ws += (size_t)NTOK * DIM;             // 2048 x 768
    float* xz    = ws;  ws += (size_t)NTOK * 2 * D_INNER;     // 2048 x 3072
    float* u     = ws;  ws += (size_t)NTOK * D_INNER;         // 2048 x 1536
    float* dbc   = ws;  ws += (size_t)NTOK * DBC_W;           // 2048 x 80
    float* delta = ws;  ws += (size_t)NTOK * D_INNER;         // 2048 x 1536
    float* yb    = ws;                                        // 2048 x 1536

    // 1) LayerNorm
    mamba_ln_kernel<<<NTOK / 8, 256, 0, stream>>>(x, ln_w, ln_b, xn);

    // 2) in_proj: xz = xn @ in_w^T     (2048 x 3072, K=768)
    mamba_gemm_wmma_f32<0><<<dim3(NTOK / 32, (2 * D_INNER) / 64), 256, 0, stream>>>(
        xn, DIM, in_w, nullptr, xz, NTOK, 2 * D_INNER, DIM);

    // 3) depthwise causal conv + SiLU -> u
    mamba_conv_silu_kernel<<<(NTOK * D_INNER) / 256, 256, 0, stream>>>(
        xz, conv_w, conv_b, u);

    // 4) x_proj: dbc = u @ xp_w^T     (2048 x 80, K=1536)
    mamba_gemm_wmma_f32<0><<<dim3(NTOK / 32, (DBC_W + 63) / 64), 256, 0, stream>>>(
        u, D_INNER, xp_w, nullptr, dbc, NTOK, DBC_W, D_INNER);

    // 5) dt_proj + softplus: delta = softplus(dbc[:, :48] @ dt_w^T + dt_b)
    mamba_gemm_wmma_f32<1><<<dim3(NTOK / 32, D_INNER / 64), 256, 0, stream>>>(
        dbc, DBC_W, dt_w, dt_b, delta, NTOK, D_INNER, DT_RANK);

    // 6) selective scan + D skip + SiLU(z) gate -> yb
    mamba_scan_kernel<<<BATCH * (D_INNER / 256), 256, 0, stream>>>(
        u, delta, dbc, xz, A_log, Dp, yb);

    // 7) out_proj: out = yb @ out_w^T  (2048 x 768, K=1536)
    mamba_gemm_wmma_f32<0><<<dim3(NTOK / 32, DIM / 64), 256, 0, stream>>>(
        yb, D_INNER, out_w, nullptr, out, NTOK, DIM, D_INNER);
}